// WeightedEmbedding_10617159156022
// MI455X (gfx1250) — compile-verified
//
#include <hip/hip_runtime.h>

// Problem constants (match reference)
#define DIM   768
#define SEQ   512
#define NW    300
#define NB    64
#define NCH   6          // 768 floats / (32 lanes * 4 floats)
#define WAVES_PER_BLOCK 8

typedef float v2f __attribute__((ext_vector_type(2)));
typedef float f4v __attribute__((ext_vector_type(4)));
typedef float v8f __attribute__((ext_vector_type(8)));

// ---------------------------------------------------------------------------
// Kernel 1: per-batch first break index: first[b] = min{w : en<S && span<=0},
// else NW.  One block per batch, LDS min-reduce.
// ---------------------------------------------------------------------------
__global__ void __launch_bounds__(256)
first_break_kernel(const int* __restrict__ wi, int* __restrict__ d_first) {
    const int b = blockIdx.x;
    const int t = threadIdx.x;
    int cand = NW;
    for (int w = t; w < NW; w += 256) {
        const int* e = wi + ((size_t)b * NW + w) * 3;
        const int st = e[1];
        const int en = e[2];
        if (en < SEQ && (en - st) <= 0) cand = min(cand, w);
    }
    __shared__ int red[256];
    red[t] = cand;
    __syncthreads();
    for (int s = 128; s > 0; s >>= 1) {
        if (t < s) red[t] = min(red[t], red[t + s]);
        __syncthreads();
    }
    if (t == 0) d_first[b] = red[0];
}

// full-wave (32 lane) sum, all lanes receive the result
__device__ __forceinline__ float wave_sum32(float v) {
    v += __shfl_xor(v, 1, 32);
    v += __shfl_xor(v, 2, 32);
    v += __shfl_xor(v, 4, 32);
    v += __shfl_xor(v, 8, 32);
    v += __shfl_xor(v, 16, 32);
    return v;
}

// ---------------------------------------------------------------------------
// Kernel 2: one wave32 per word.
// ---------------------------------------------------------------------------
__global__ void __launch_bounds__(256)
weighted_emb_kernel(const float* __restrict__ ernie,
                    const float* __restrict__ emb,
                    const int*   __restrict__ wi,
                    const int*   __restrict__ d_first,
                    float*       __restrict__ out) {
    const int lane = threadIdx.x & 31;
    const int wave = threadIdx.x >> 5;
    const int word = blockIdx.x * WAVES_PER_BLOCK + wave;   // grid covers exactly NB*NW
    const int b    = word / NW;
    const int w    = word - b * NW;

    const int* e  = wi + ((size_t)b * NW + w) * 3;
    const int wid = e[0];
    const int st  = e[1];
    const int en  = e[2];
    const int first = d_first[b];

    f4v* dst = (f4v*)(out + (size_t)word * DIM);

    // ---- fill path: w >= first break ----
    if (w >= first) {
        const int fw  = first < (NW - 1) ? first : (NW - 1);
        const int fid = wi[((size_t)b * NW + fw) * 3];
        const f4v* src = (const f4v*)(emb + (size_t)fid * DIM);
#pragma unroll
        for (int c = 0; c < NCH; ++c) {
            const int idx = c * 32 + lane;
            __builtin_nontemporal_store(src[idx], dst + idx);
        }
        return;
    }

    // ---- out-of-range path: just the word embedding ----
    if (en >= SEQ) {
        const f4v* src = (const f4v*)(emb + (size_t)wid * DIM);
#pragma unroll
        for (int c = 0; c < NCH; ++c) {
            const int idx = c * 32 + lane;
            __builtin_nontemporal_store(src[idx], dst + idx);
        }
        return;
    }

    // ---- attention path (wave-uniform: EXEC is all ones here) ----
    const f4v* wrow = (const f4v*)(emb + (size_t)wid * DIM);
    const f4v* crow[4];
#pragma unroll
    for (int l = 0; l < 4; ++l) {
        int pos = st + l;
        pos = pos < 0 ? 0 : pos;
        pos = pos > (SEQ - 1) ? (SEQ - 1) : pos;
        crow[l] = (const f4v*)(ernie + ((size_t)b * SEQ + pos) * DIM);
    }

    // Pass 1: per-lane partial dot products  p[l] = sum_d we[d]*chars[l][d]
    float p0 = 0.f, p1 = 0.f, p2 = 0.f, p3 = 0.f;
#pragma unroll
    for (int c = 0; c < NCH; ++c) {
        const int idx = c * 32 + lane;
        const f4v wv = wrow[idx];
        const f4v c0 = crow[0][idx];
        const f4v c1 = crow[1][idx];
        const f4v c2 = crow[2][idx];
        const f4v c3 = crow[3][idx];
        p0 = fmaf(wv.x, c0.x, fmaf(wv.y, c0.y, fmaf(wv.z, c0.z, fmaf(wv.w, c0.w, p0))));
        p1 = fmaf(wv.x, c1.x, fmaf(wv.y, c1.y, fmaf(wv.z, c1.z, fmaf(wv.w, c1.w, p1))));
        p2 = fmaf(wv.x, c2.x, fmaf(wv.y, c2.y, fmaf(wv.z, c2.z, fmaf(wv.w, c2.w, p2))));
        p3 = fmaf(wv.x, c3.x, fmaf(wv.y, c3.y, fmaf(wv.z, c3.z, fmaf(wv.w, c3.w, p3))));
    }

    // WMMA-assisted cross-lane reduction (fp32-exact).
    // A (16x4) is a constant selector: row0 = (1,0,1,0), row1 = (0,1,0,1),
    // other rows 0.  B (4x16) holds per-lane partials with zero marshaling:
    //   lanes 0-15 supply K=0,1 of column N=lane; lanes 16-31 supply K=2,3.
    // => D[0,n] = p_n[l0] + p_{n+16}[l0],  D[1,n] = same for l1.
    v2f av;
    av.x = ((lane & 15) == 0) ? 1.0f : 0.0f;
    av.y = ((lane & 15) == 1) ? 1.0f : 0.0f;
    v8f cz = {0.f, 0.f, 0.f, 0.f, 0.f, 0.f, 0.f, 0.f};
    v2f b01; b01.x = p0; b01.y = p1;
    v2f b23; b23.x = p2; b23.y = p3;
    v8f q01 = __builtin_amdgcn_wmma_f32_16x16x4_f32(false, av, false, b01,
                                                    (short)0, cz, false, false);
    v8f q23 = __builtin_amdgcn_wmma_f32_16x16x4_f32(false, av, false, b23,
                                                    (short)0, cz, false, false);
    // rows live in element 0/1 of lanes 0-15 (lanes 16-31 carry zero rows);
    // a full 32-lane sum finishes the reduction and broadcasts everywhere.
    float s[4];
    s[0] = wave_sum32(q01[0]);
    s[1] = wave_sum32(q01[1]);
    s[2] = wave_sum32(q23[0]);
    s[3] = wave_sum32(q23[1]);

    // masked softmax over l < max(span,1)
    const int span  = en - st;
    const int spanm = span < 1 ? 1 : span;
    float m = s[0];
#pragma unroll
    for (int l = 1; l < 4; ++l)
        if (l < spanm) m = fmaxf(m, s[l]);
    float a[4];
    float esum = 0.f;
#pragma unroll
    for (int l = 0; l < 4; ++l) {
        a[l] = (l < spanm) ? __expf(s[l] - m) : 0.0f;
        esum += a[l];
    }
    const float inv = 1.0f / esum;
#pragma unroll
    for (int l = 0; l < 4; ++l) a[l] *= inv;

    // Pass 2: pooled[d] = sum_l a[l] * chars[l][d]  (char rows hit WGP$)
#pragma unroll
    for (int c = 0; c < NCH; ++c) {
        const int idx = c * 32 + lane;
        const f4v c0 = crow[0][idx];
        const f4v c1 = crow[1][idx];
        const f4v c2 = crow[2][idx];
        const f4v c3 = crow[3][idx];
        f4v acc;
        acc.x = fmaf(a[0], c0.x, fmaf(a[1], c1.x, fmaf(a[2], c2.x, a[3] * c3.x)));
        acc.y = fmaf(a[0], c0.y, fmaf(a[1], c1.y, fmaf(a[2], c2.y, a[3] * c3.y)));
        acc.z = fmaf(a[0], c0.z, fmaf(a[1], c1.z, fmaf(a[2], c2.z, a[3] * c3.z)));
        acc.w = fmaf(a[0], c0.w, fmaf(a[1], c1.w, fmaf(a[2], c2.w, a[3] * c3.w)));
        __builtin_nontemporal_store(acc, dst + idx);
    }
}

extern "C" void kernel_launch(void* const* d_in, const int* in_sizes, int n_in,
                              void* d_out, int out_size, void* d_ws, size_t ws_size,
                              hipStream_t stream) {
    const float* ernie = (const float*)d_in[0];   // (B,S,D) fp32
    const float* emb   = (const float*)d_in[1];   // (VOCAB,D) fp32
    const int*   wi    = (const int*)d_in[2];     // (B,W,3) int32
    float* out = (float*)d_out;                   // (B,W,D) fp32
    int* d_first = (int*)d_ws;                    // NB ints of scratch

    first_break_kernel<<<NB, 256, 0, stream>>>(wi, d_first);

    const int words  = NB * NW;                   // 19200, divisible by 8
    const int blocks = words / WAVES_PER_BLOCK;   // 2400
    weighted_emb_kernel<<<blocks, 256, 0, stream>>>(ernie, emb, wi, d_first, out);
}